// IPALayer_27788438405631
// MI455X (gfx1250) — compile-verified
//
#include <hip/hip_runtime.h>
#include <hip/hip_bf16.h>
#include <math.h>

// ---------------------------------------------------------------------------
// Problem constants (from the reference)
// ---------------------------------------------------------------------------
constexpr int N_ = 4, L_ = 1024, HID_ = 128, H_ = 12;
constexpr int DK_ = 16, DV_ = 16, PQK_ = 4, PV_ = 8;
constexpr int M_ = N_ * L_;                 // 4096 tokens
constexpr int QKDIM = H_ * DK_;             // 192
constexpr int VDIM  = H_ * DV_;             // 192
constexpr int QVDIM = H_ * PQK_ * 3;        // 144
constexpr int VVDIM = H_ * PV_ * 3;         // 288
constexpr int FEATD = H_ * (DV_ + PV_ * 4); // 576

typedef __attribute__((ext_vector_type(2))) float v2f;
typedef __attribute__((ext_vector_type(8))) float v8f;

// D = A(16x4) * B(4x16) + C  (dense f32 WMMA, CDNA5)
__device__ __forceinline__ v8f wmma_k4(v2f a, v2f b, v8f c) {
  return __builtin_amdgcn_wmma_f32_16x16x4_f32(
      /*neg_a=*/false, a, /*neg_b=*/false, b,
      /*c_mod=*/(short)0, c, /*reuse_a=*/false, /*reuse_b=*/false);
}

__device__ __forceinline__ float rmax16(float x) {
#pragma unroll
  for (int o = 1; o < 16; o <<= 1) x = fmaxf(x, __shfl_xor(x, o, 16));
  return x;
}
__device__ __forceinline__ float rsum16(float x) {
#pragma unroll
  for (int o = 1; o < 16; o <<= 1) x += __shfl_xor(x, o, 16);
  return x;
}

// ---------------------------------------------------------------------------
// Generic row-major f32 GEMM: C(MxNo) = A(MxK) @ B(KxNo) [+ bias]
// One wave (32 threads) computes one 16x16 tile via chained v_wmma_f32_16x16x4.
// M % 16 == 0, No % 16 == 0, K % 4 == 0.
// ---------------------------------------------------------------------------
__global__ __launch_bounds__(32)
void gemm16_wmma(const float* __restrict__ A, const float* __restrict__ B,
                 const float* __restrict__ bias, float* __restrict__ C,
                 int K, int No) {
  const int m0 = blockIdx.x * 16, n0 = blockIdx.y * 16;
  const int lane = threadIdx.x, half = lane >> 4, lid = lane & 15;
  const float* arow = A + (size_t)(m0 + lid) * K;
  v8f acc = {};
  for (int k0 = 0; k0 < K; k0 += 4) {
    const int kk = k0 + 2 * half;
    v2f a, b;
    a.x = arow[kk];               a.y = arow[kk + 1];
    b.x = B[(size_t)kk * No + n0 + lid];
    b.y = B[(size_t)(kk + 1) * No + n0 + lid];
    acc = wmma_k4(a, b, acc);
  }
  const float bb = bias ? bias[n0 + lid] : 0.0f;
#pragma unroll
  for (int r = 0; r < 8; ++r) {
    const int row = m0 + half * 8 + r;
    C[(size_t)row * No + n0 + lid] = acc[r] + bb;
  }
}

// ---------------------------------------------------------------------------
// Frame apply: out_g[n,h,l,p*3+x] = R[n,l] @ vec[n,l,h,p] + t[n,l]
// optional out_sq[n,h,l] = sum_{p,x} g^2
// ---------------------------------------------------------------------------
__global__ void frame_apply(const float* __restrict__ vec,
                            const float* __restrict__ R,
                            const float* __restrict__ t,
                            float* __restrict__ outg,
                            float* __restrict__ outsq, int P) {
  int tid = blockIdx.x * blockDim.x + threadIdx.x;
  if (tid >= N_ * L_ * H_) return;
  const int h = tid % H_;
  const int l = (tid / H_) % L_;
  const int n = tid / (H_ * L_);
  const float* Rm = R + (size_t)(n * L_ + l) * 9;
  const float* tv = t + (size_t)(n * L_ + l) * 3;
  const float* vin = vec + (size_t)(n * L_ + l) * (H_ * P * 3) + (size_t)h * P * 3;
  float* go = outg + ((size_t)(n * H_ + h) * L_ + l) * (P * 3);
  float sq = 0.0f;
  for (int p = 0; p < P; ++p) {
    const float vx = vin[p * 3 + 0], vy = vin[p * 3 + 1], vz = vin[p * 3 + 2];
#pragma unroll
    for (int x = 0; x < 3; ++x) {
      const float g = Rm[x * 3 + 0] * vx + Rm[x * 3 + 1] * vy + Rm[x * 3 + 2] * vz + tv[x];
      go[p * 3 + x] = g;
      sq += g * g;
    }
  }
  if (outsq) outsq[(size_t)(n * H_ + h) * L_ + l] = sq;
}

// ---------------------------------------------------------------------------
// Flash-style IPA attention. One wave per (n, h, 16-query tile).
//   logits = c1*(q.k) + sA*(qg.kg) - c2*(q2+k2)   (scales folded into A-frags)
//   online softmax over keys; accumulate o (16 wide) and o_vec (24 wide).
// P tile goes through LDS to convert C-layout -> A-layout.
// ---------------------------------------------------------------------------
__global__ __launch_bounds__(32)
void ipa_attn(const float* __restrict__ q,  const float* __restrict__ k,
              const float* __restrict__ v,  const float* __restrict__ qg,
              const float* __restrict__ kg, const float* __restrict__ q2,
              const float* __restrict__ k2, const float* __restrict__ vg,
              const unsigned char* __restrict__ mask,
              const float* __restrict__ gamma,
              float* __restrict__ feat, float* __restrict__ ovec) {
  const int it = blockIdx.x, h = blockIdx.y, n = blockIdx.z;
  const int i0 = it * 16;
  const int lane = threadIdx.x, half = lane >> 4, lid = lane & 15;

  const float sp = log1pf(__expf(gamma[h]));          // softplus(gamma_h)
  const float c1 = 0.70710678f * 0.25f;                // 2^-1/2 * DK^-1/2
  const float sA = 0.70710678f * sp;                   // 2^-1/2 * sp * PQK^-1/2 * 2
  const float c2 = 0.35355339f * sp;                   // 2^-1/2 * sp * PQK^-1/2

  // Pre-load A-fragments: queries (K=16 -> 4 chunks) and query points (K=12 -> 3)
  const float* qrow  = q  + ((size_t)(n * L_ + i0 + lid)) * QKDIM + h * DK_;
  const float* qgrow = qg + ((size_t)(n * H_ + h) * L_ + i0 + lid) * 12;
  v2f aq[4], ag[3];
#pragma unroll
  for (int c = 0; c < 4; ++c) {
    const int kk = c * 4 + 2 * half;
    aq[c].x = qrow[kk] * c1; aq[c].y = qrow[kk + 1] * c1;
  }
#pragma unroll
  for (int c = 0; c < 3; ++c) {
    const int kk = c * 4 + 2 * half;
    ag[c].x = qgrow[kk] * sA; ag[c].y = qgrow[kk + 1] * sA;
  }
  float q2r[8];
#pragma unroll
  for (int r = 0; r < 8; ++r)
    q2r[r] = q2[(size_t)(n * H_ + h) * L_ + i0 + half * 8 + r];

  float mrow[8], lrow[8];
#pragma unroll
  for (int r = 0; r < 8; ++r) { mrow[r] = -INFINITY; lrow[r] = 0.0f; }
  v8f acc_o = {}, acc_v0 = {}, acc_v1 = {};

  __shared__ float P[16][17];

  for (int jt = 0; jt < L_; jt += 16) {
    // ---- logits tile: S = c1*Q.K^T + sA*QG.KG^T (7 chained WMMAs) ----
    const float* krow  = k  + ((size_t)(n * L_ + jt + lid)) * QKDIM + h * DK_;
    const float* kgrow = kg + ((size_t)(n * H_ + h) * L_ + jt + lid) * 12;
    v8f s = {};
#pragma unroll
    for (int c = 0; c < 4; ++c) {
      const int kk = c * 4 + 2 * half;
      v2f b; b.x = krow[kk]; b.y = krow[kk + 1];
      s = wmma_k4(aq[c], b, s);
    }
#pragma unroll
    for (int c = 0; c < 3; ++c) {
      const int kk = c * 4 + 2 * half;
      v2f b; b.x = kgrow[kk]; b.y = kgrow[kk + 1];
      s = wmma_k4(ag[c], b, s);
    }
    const int gj = jt + lid;
    const float k2c = k2[(size_t)(n * H_ + h) * L_ + gj];
    const bool mok = mask[(size_t)n * L_ + gj] != 0;
    if (jt + 16 < L_)  // prefetch next key tile
      __builtin_prefetch(k + ((size_t)(n * L_ + jt + 16 + lid)) * QKDIM + h * DK_, 0, 3);

    // ---- online softmax (rows live in VGPR index, cols across 16 lanes) ----
#pragma unroll
    for (int r = 0; r < 8; ++r) {
      const int gi = i0 + half * 8 + r;
      float val = s[r] - c2 * (q2r[r] + k2c);
      if (gi == gj) val = -100.0f;
      if (!mok)     val = -INFINITY;
      const float tmax = rmax16(val);
      const float mnew = fmaxf(mrow[r], tmax);
      const float corr = __expf(mrow[r] - mnew);   // 0 on first tile
      mrow[r] = mnew;
      const float pe = __expf(val - mnew);
      lrow[r] = lrow[r] * corr + rsum16(pe);
      acc_o[r]  *= corr;
      acc_v0[r] *= corr;
      acc_v1[r] *= corr;
      P[half * 8 + r][lid] = pe;                   // C-layout -> LDS
    }
    __syncthreads();

    // ---- accumulate O += P@V (16 wide), O_vec += P@VG (24 wide) ----
#pragma unroll
    for (int c = 0; c < 4; ++c) {
      const int kk = c * 4 + 2 * half;             // key index within tile
      v2f ap; ap.x = P[lid][kk]; ap.y = P[lid][kk + 1];   // A-layout read
      v2f bv;
      bv.x = v[((size_t)(n * L_ + jt + kk))     * VDIM + h * DV_ + lid];
      bv.y = v[((size_t)(n * L_ + jt + kk + 1)) * VDIM + h * DV_ + lid];
      acc_o = wmma_k4(ap, bv, acc_o);
      const float* vgr = vg + ((size_t)(n * H_ + h) * L_ + jt + kk) * 24;
      v2f b0; b0.x = vgr[lid];       b0.y = vgr[24 + lid];
      acc_v0 = wmma_k4(ap, b0, acc_v0);
      const int c2i = 16 + lid;
      v2f b1;
      b1.x = (c2i < 24) ? vgr[c2i]      : 0.0f;
      b1.y = (c2i < 24) ? vgr[24 + c2i] : 0.0f;
      acc_v1 = wmma_k4(ap, b1, acc_v1);
    }
    __syncthreads();
  }

  // ---- normalize and store: o -> feat[:, h*16+lid], o_vec -> ws ----
#pragma unroll
  for (int r = 0; r < 8; ++r) {
    const int gi = i0 + half * 8 + r;
    const float inv = (lrow[r] > 0.0f) ? (1.0f / lrow[r]) : 0.0f;
    feat[(size_t)(n * L_ + gi) * FEATD + h * DV_ + lid] = acc_o[r] * inv;
    float* ov = ovec + ((size_t)(n * L_ + gi) * H_ + h) * 24;
    ov[lid] = acc_v0[r] * inv;
    if (16 + lid < 24) ov[16 + lid] = acc_v1[r] * inv;
  }
}

// ---------------------------------------------------------------------------
// o_loc = R^T (o_vec - t); norm; write into feat[192:576)
// ---------------------------------------------------------------------------
__global__ void finalize_feat(const float* __restrict__ ovec,
                              const float* __restrict__ R,
                              const float* __restrict__ t,
                              float* __restrict__ feat) {
  int tid = blockIdx.x * blockDim.x + threadIdx.x;
  if (tid >= N_ * L_ * H_ * PV_) return;
  const int m = tid % PV_;
  const int h = (tid / PV_) % H_;
  const int l = (tid / (PV_ * H_)) % L_;
  const int n = tid / (PV_ * H_ * L_);
  const float* Rm = R + (size_t)(n * L_ + l) * 9;
  const float* tv = t + (size_t)(n * L_ + l) * 3;
  const float* ov = ovec + ((size_t)(n * L_ + l) * H_ + h) * 24 + m * 3;
  const float dx = ov[0] - tv[0], dy = ov[1] - tv[1], dz = ov[2] - tv[2];
  float* fb = feat + (size_t)(n * L_ + l) * FEATD;
  float n2 = 0.0f;
#pragma unroll
  for (int y = 0; y < 3; ++y) {
    const float loc = Rm[0 * 3 + y] * dx + Rm[1 * 3 + y] * dy + Rm[2 * 3 + y] * dz;
    fb[H_ * DV_ + (h * PV_ + m) * 3 + y] = loc;
    n2 += loc * loc;
  }
  fb[H_ * DV_ + H_ * PV_ * 3 + h * PV_ + m] = sqrtf(n2);
}

// ---------------------------------------------------------------------------
extern "C" void kernel_launch(void* const* d_in, const int* in_sizes, int n_in,
                              void* d_out, int out_size, void* d_ws, size_t ws_size,
                              hipStream_t stream) {
  const float* x    = (const float*)d_in[0];
  const float* R    = (const float*)d_in[1];
  const float* t    = (const float*)d_in[2];
  const unsigned char* mask = (const unsigned char*)d_in[3];
  const float* Wq   = (const float*)d_in[4];
  const float* Wk   = (const float*)d_in[5];
  const float* Wv   = (const float*)d_in[6];
  const float* Wqv  = (const float*)d_in[7];
  const float* Wkv  = (const float*)d_in[8];
  const float* Wvv  = (const float*)d_in[9];
  const float* gamma= (const float*)d_in[10];
  const float* Wo   = (const float*)d_in[11];
  const float* bo   = (const float*)d_in[12];
  float* out = (float*)d_out;

  // workspace layout (floats); raw-projection buffer is reused as feat
  float* ws = (float*)d_ws;
  size_t off = 0;
  float* q    = ws + off; off += (size_t)M_ * QKDIM;       // 786432
  float* k    = ws + off; off += (size_t)M_ * QKDIM;
  float* v    = ws + off; off += (size_t)M_ * VDIM;
  float* raw  = ws + off; off += (size_t)M_ * FEATD;       // qv|kv|vv, later feat
  float* qg   = ws + off; off += (size_t)N_ * H_ * L_ * 12;
  float* kg   = ws + off; off += (size_t)N_ * H_ * L_ * 12;
  float* q2   = ws + off; off += (size_t)N_ * H_ * L_;
  float* k2   = ws + off; off += (size_t)N_ * H_ * L_;
  float* vg   = ws + off; off += (size_t)N_ * H_ * L_ * 24;
  float* ovec = ws + off; off += (size_t)M_ * VVDIM;
  float* qvr = raw;
  float* kvr = raw + (size_t)M_ * QVDIM;
  float* vvr = raw + (size_t)M_ * QVDIM * 2;
  float* feat = raw;

  const dim3 blk(32);
  // 1) projections (f32 WMMA GEMMs)
  gemm16_wmma<<<dim3(M_ / 16, QKDIM / 16), blk, 0, stream>>>(x, Wq, nullptr, q,  HID_, QKDIM);
  gemm16_wmma<<<dim3(M_ / 16, QKDIM / 16), blk, 0, stream>>>(x, Wk, nullptr, k,  HID_, QKDIM);
  gemm16_wmma<<<dim3(M_ / 16, VDIM  / 16), blk, 0, stream>>>(x, Wv, nullptr, v,  HID_, VDIM);
  gemm16_wmma<<<dim3(M_ / 16, QVDIM / 16), blk, 0, stream>>>(x, Wqv, nullptr, qvr, HID_, QVDIM);
  gemm16_wmma<<<dim3(M_ / 16, QVDIM / 16), blk, 0, stream>>>(x, Wkv, nullptr, kvr, HID_, QVDIM);
  gemm16_wmma<<<dim3(M_ / 16, VVDIM / 16), blk, 0, stream>>>(x, Wvv, nullptr, vvr, HID_, VVDIM);

  // 2) frame-apply
  const int nfh = N_ * L_ * H_;
  frame_apply<<<(nfh + 255) / 256, 256, 0, stream>>>(qvr, R, t, qg, q2, PQK_);
  frame_apply<<<(nfh + 255) / 256, 256, 0, stream>>>(kvr, R, t, kg, k2, PQK_);
  frame_apply<<<(nfh + 255) / 256, 256, 0, stream>>>(vvr, R, t, vg, nullptr, PV_);

  // 3) attention (overwrites raw buffer's o-region of feat)
  ipa_attn<<<dim3(L_ / 16, H_, N_), blk, 0, stream>>>(q, k, v, qg, kg, q2, k2, vg,
                                                      mask, gamma, feat, ovec);

  // 4) inverse frame + norm into feat
  const int nfin = N_ * L_ * H_ * PV_;
  finalize_feat<<<(nfin + 255) / 256, 256, 0, stream>>>(ovec, R, t, feat);

  // 5) output GEMM with bias
  gemm16_wmma<<<dim3(M_ / 16, HID_ / 16), blk, 0, stream>>>(feat, Wo, bo, out, FEATD, HID_);

  (void)in_sizes; (void)n_in; (void)out_size; (void)ws_size;
}